// SpGraphAttentionLayer_48550310314069
// MI455X (gfx1250) — compile-verified
//
#include <hip/hip_runtime.h>
#include <hip/hip_bf16.h>

#define DIM 128
#define ALPHA 0.2f
#define MTILES 5   // M-tiles (16 rows each) swept per block; 50000/16/5 = 625 blocks

typedef __attribute__((ext_vector_type(2)))  float    v2f;
typedef __attribute__((ext_vector_type(8)))  float    v8f;
typedef __attribute__((ext_vector_type(16))) _Float16 v16h;

// ---------------------------------------------------------------------------
// Kernel 1: H = X @ W   (N x 128) = (N x 128) x (128 x 128), fp32 via WMMA.
// One wave owns one 16-wide N-slice (8 waves cover all 128 cols) and keeps all
// 32 B-fragments (64 VGPRs) resident while sweeping MTILES M-tiles, so W is
// fetched from L2 once per block instead of once per tile.
// ---------------------------------------------------------------------------
__global__ __launch_bounds__(256) void gat_gemm(const float* __restrict__ X,
                                                const float* __restrict__ W,
                                                float* __restrict__ H, int N) {
    __shared__ float xs[16 * DIM];  // 8 KB staging for one 16x128 X slab
    const int tid  = threadIdx.x;
    const int wave = tid >> 5;
    const int lane = tid & 31;
    const int n0   = wave * 16;
    const int half = (lane < 16) ? 0 : 1;  // lane half selects K sub-pair
    const int l15  = lane & 15;

    // Preload all B fragments for this wave's 16 columns: B 4x16 layout,
    // v0 <-> K = (half?2:0), v1 <-> K+1, lane&15 = N column.
    v2f bf[32];
#pragma unroll
    for (int kk = 0; kk < 32; ++kk) {
        const int kb = kk * 4 + half * 2;
        bf[kk].x = W[kb * DIM + n0 + l15];
        bf[kk].y = W[(kb + 1) * DIM + n0 + l15];
    }

    for (int t = 0; t < MTILES; ++t) {
        const int m0 = (blockIdx.x * MTILES + t) * 16;
        if (m0 + 16 > N) return;  // uniform over block

        __syncthreads();  // previous tile's readers done before overwrite
        // Cooperative copy of 16 contiguous rows (2048 floats) into LDS.
        const float4* gsrc = reinterpret_cast<const float4*>(X + (size_t)m0 * DIM);
        float4*       ldst = reinterpret_cast<float4*>(xs);
        ldst[tid]       = gsrc[tid];
        ldst[tid + 256] = gsrc[tid + 256];
        __syncthreads();

        v8f c = {};
#pragma unroll
        for (int kk = 0; kk < 32; ++kk) {
            // A 16x4 layout: v0 <-> K = (half?2:0), v1 <-> K+1, lane&15 = M row.
            const int kb = kk * 4 + half * 2;
            v2f a;
            a.x = xs[l15 * DIM + kb];
            a.y = xs[l15 * DIM + kb + 1];
            c = __builtin_amdgcn_wmma_f32_16x16x4_f32(false, a, false, bf[kk],
                                                      (short)0, c, false, false);
        }

        // C/D layout: VGPR r -> M = r + (half?8:0), N = lane&15.
        const int nCol  = n0 + l15;
        const int mBase = m0 + half * 8;
#pragma unroll
        for (int r = 0; r < 8; ++r)
            H[(size_t)(mBase + r) * DIM + nCol] = c[r];
    }
}

// ---------------------------------------------------------------------------
// Kernel 2: s1[n] = H[n,:].a[:128], s2[n] = H[n,:].a[128:]. One wave / node.
// ---------------------------------------------------------------------------
__global__ __launch_bounds__(256) void gat_scores(const float* __restrict__ H,
                                                  const float* __restrict__ a,
                                                  float* __restrict__ s1,
                                                  float* __restrict__ s2, int N) {
    const int lane = threadIdx.x & 31;
    const int node = blockIdx.x * 8 + (threadIdx.x >> 5);
    if (node >= N) return;
    const float4 hv = reinterpret_cast<const float4*>(H + (size_t)node * DIM)[lane];
    const float4 a1 = reinterpret_cast<const float4*>(a)[lane];
    const float4 a2 = reinterpret_cast<const float4*>(a + DIM)[lane];
    float d1 = hv.x * a1.x + hv.y * a1.y + hv.z * a1.z + hv.w * a1.w;
    float d2 = hv.x * a2.x + hv.y * a2.y + hv.z * a2.z + hv.w * a2.w;
    for (int off = 16; off > 0; off >>= 1) {
        d1 += __shfl_xor(d1, off, 32);
        d2 += __shfl_xor(d2, off, 32);
    }
    if (lane == 0) { s1[node] = d1; s2[node] = d2; }
}

// ---------------------------------------------------------------------------
// Kernel 3: per-edge exp(leakyrelu(s1[src]+s2[dst])), atomic scatter into
// rowsum[src] and hp[src,:] += w * H[dst,:]. One wave / edge; lane -> float4
// slice of the 128-wide row. H is L2-resident (25.6 MB < 192 MB L2).
// ---------------------------------------------------------------------------
__global__ __launch_bounds__(256) void gat_edges(const long long* __restrict__ edge,
                                                 const float* __restrict__ H,
                                                 const float* __restrict__ s1,
                                                 const float* __restrict__ s2,
                                                 float* __restrict__ hp,
                                                 float* __restrict__ rowsum,
                                                 int E) {
    const int lane   = threadIdx.x & 31;
    const int nwaves = gridDim.x * 8;
    for (int e = blockIdx.x * 8 + (threadIdx.x >> 5); e < E; e += nwaves) {
        const int s = (int)edge[e];
        const int d = (int)edge[(size_t)E + e];
        float sc = s1[s] + s2[d];
        sc = (sc > 0.f) ? sc : ALPHA * sc;
        const float w = expf(sc);
        if (lane == 0) unsafeAtomicAdd(&rowsum[s], w);
        const float4 hv = reinterpret_cast<const float4*>(H + (size_t)d * DIM)[lane];
        float* o = hp + (size_t)s * DIM + lane * 4;
        unsafeAtomicAdd(o + 0, w * hv.x);
        unsafeAtomicAdd(o + 1, w * hv.y);
        unsafeAtomicAdd(o + 2, w * hv.z);
        unsafeAtomicAdd(o + 3, w * hv.w);
    }
}

// ---------------------------------------------------------------------------
// Kernel 4: out = elu(hp / rowsum)
// ---------------------------------------------------------------------------
__global__ __launch_bounds__(256) void gat_final(const float* __restrict__ hp,
                                                 const float* __restrict__ rowsum,
                                                 float* __restrict__ out,
                                                 long long total) {
    const long long i = (long long)blockIdx.x * 256 + threadIdx.x;
    if (i >= total) return;
    const float x = hp[i] / rowsum[i / DIM];
    out[i] = (x > 0.f) ? x : (expf(x) - 1.f);
}

extern "C" void kernel_launch(void* const* d_in, const int* in_sizes, int n_in,
                              void* d_out, int out_size, void* d_ws, size_t ws_size,
                              hipStream_t stream) {
    const float*     X    = (const float*)d_in[0];      // [N,128]
    const long long* edge = (const long long*)d_in[1];  // [2,E] int64
    const float*     W    = (const float*)d_in[2];      // [128,128]
    const float*     a    = (const float*)d_in[3];      // [256]
    const int N = in_sizes[0] / DIM;
    const int E = in_sizes[1] / 2;

    // Workspace: H | s1 | s2 | rowsum | hp  (rowsum & hp contiguous for memset)
    float* ws     = (float*)d_ws;
    float* H      = ws;
    float* s1     = H + (size_t)N * DIM;
    float* s2     = s1 + N;
    float* rowsum = s2 + N;
    float* hp     = rowsum + N;

    hipMemsetAsync(rowsum, 0, sizeof(float) * ((size_t)N * DIM + N), stream);

    const int mtiles = (N + 15) / 16;
    gat_gemm  <<<(mtiles + MTILES - 1) / MTILES, 256, 0, stream>>>(X, W, H, N);
    gat_scores<<<(N + 7) / 8, 256, 0, stream>>>(H, a, s1, s2, N);
    gat_edges <<<4096, 256, 0, stream>>>(edge, H, s1, s2, hp, rowsum, E);

    const long long total = (long long)N * DIM;
    gat_final <<<(int)((total + 255) / 256), 256, 0, stream>>>(hp, rowsum,
                                                               (float*)d_out, total);
}